// SWaVStruct_41790031790765
// MI455X (gfx1250) — compile-verified
//
#include <hip/hip_runtime.h>

typedef float v2f __attribute__((ext_vector_type(2)));
typedef float v8f __attribute__((ext_vector_type(8)));

namespace {
constexpr int   BATCH = 256;
constexpr int   PATCH = 400;
constexpr int   KCLS  = 12;
constexpr int   EMB   = 256;
constexpr int   BP    = 800;         // 2*PATCH rows in sinkhorn matrix
constexpr int   NEL   = BP * KCLS;   // 9600
constexpr int   NT    = 256;         // threads per block (8 waves)
constexpr int   NWAVE = NT / 32;
constexpr float TEMP  = 0.1f;
constexpr float EPS   = 0.05f;
}

__global__ __launch_bounds__(NT)
void swav_sample_kernel(const float* __restrict__ scores_t,
                        const float* __restrict__ scores_s,
                        const float* __restrict__ queue_base,
                        const float* __restrict__ queue_mod,
                        const float* __restrict__ proto_w,
                        float* __restrict__ loss_ws)
{
    __shared__ float pwL[16 * EMB];      // proto_w padded to 16 rows (rows 12..15 = 0)
    __shared__ float M[NEL];             // sinkhorn matrix [800][12]
    __shared__ float red[NWAVE * KCLS];  // cross-wave reduction scratch
    __shared__ float bc[16];             // broadcast scratch

    const int tid  = threadIdx.x;
    const int lane = tid & 31;
    const int wid  = tid >> 5;
    const int b    = blockIdx.x;

    // Stage proto_w [12,256] into LDS, zero-pad to 16 rows (WMMA N padding).
    for (int i = tid; i < 16 * EMB; i += NT)
        pwL[i] = (i < KCLS * EMB) ? proto_w[i] : 0.0f;
    __syncthreads();

    // WMMA f32 16x16x4 lane mapping (ISA 7.12.2):
    //   A: lane L in 0..15 -> row L, K = {0,1}; lane L+16 -> row L, K = {2,3}
    //   B: lane N in 0..15 -> col N, K = {0,1}; lane N+16 -> col N, K = {2,3}
    const int mrow = lane & 15;
    const int khi  = (lane >> 4) << 1;   // 0 or 2

    float lossAcc = 0.0f;

    for (int br = 0; br < 2; ++br) {
        const float* Qm  = (br == 0 ? queue_base : queue_mod) + (size_t)b * PATCH * EMB;
        const float* Ssf = (br == 0 ? scores_t  : scores_s ) + (size_t)b * PATCH * KCLS;
        const float* Sot = (br == 0 ? scores_s  : scores_t ) + (size_t)b * PATCH * KCLS;

        // -------- queue scores GEMM via V_WMMA_F32_16X16X4_F32: M rows 0..399 --------
        for (int tile = wid; tile < PATCH / 16; tile += NWAVE) {
            const float* Arow = Qm + (size_t)(tile * 16 + mrow) * EMB;
            if (tile + NWAVE < PATCH / 16)
                __builtin_prefetch(Qm + (size_t)((tile + NWAVE) * 16 + mrow) * EMB, 0, 0);
            v8f acc = {0.f, 0.f, 0.f, 0.f, 0.f, 0.f, 0.f, 0.f};
            #pragma unroll 4
            for (int k = 0; k < EMB; k += 4) {
                v2f a  = *(const v2f*)(Arow + k + khi);
                v2f bb = *(const v2f*)(&pwL[mrow * EMB + k + khi]); // B[k][n] = pw[n][k]
                acc = __builtin_amdgcn_wmma_f32_16x16x4_f32(
                        false, a, false, bb, (short)0, acc, false, false);
            }
            // D layout: VGPR i -> row (tile*16 + i + 8*(lane>>4)), col = lane&15
            if (mrow < KCLS) {
                const int rbase = tile * 16 + ((lane >> 4) << 3);
                #pragma unroll
                for (int i = 0; i < 8; ++i)
                    M[(rbase + i) * KCLS + mrow] = acc[i];
            }
        }

        // Rows 400..799: raw current-view scores.
        for (int i = tid; i < PATCH * KCLS; i += NT)
            M[PATCH * KCLS + i] = Ssf[i];
        __syncthreads();

        // Global max (stabilizer; the exp(-max/eps) factor cancels exactly in
        // the first class normalization — avoids f32 overflow of exp(s/0.05)).
        float mx = -3.402823466e38f;
        for (int i = tid; i < NEL; i += NT) mx = fmaxf(mx, M[i]);
        for (int o = 16; o > 0; o >>= 1) mx = fmaxf(mx, __shfl_xor(mx, o, 32));
        if (lane == 0) red[wid] = mx;
        __syncthreads();
        if (tid == 0) {
            float m = red[0];
            for (int w = 1; w < NWAVE; ++w) m = fmaxf(m, red[w]);
            bc[0] = m;
        }
        __syncthreads();
        mx = bc[0];
        for (int i = tid; i < NEL; i += NT)
            M[i] = expf((M[i] - mx) * (1.0f / EPS));
        __syncthreads();

        // 3 sinkhorn iterations. (Q/sum(Q) omitted: a global scale cancels in
        // the first class-sum division — mathematically identical.)
        for (int it = 0; it < 3; ++it) {
            float cs[KCLS];
            #pragma unroll
            for (int k = 0; k < KCLS; ++k) cs[k] = 0.0f;
            for (int r = tid; r < BP; r += NT) {
                #pragma unroll
                for (int k = 0; k < KCLS; ++k) cs[k] += M[r * KCLS + k];
            }
            #pragma unroll
            for (int k = 0; k < KCLS; ++k)
                for (int o = 16; o > 0; o >>= 1) cs[k] += __shfl_xor(cs[k], o, 32);
            if (lane == 0) {
                for (int k = 0; k < KCLS; ++k) red[wid * KCLS + k] = cs[k];
            }
            __syncthreads();
            if (tid < KCLS) {
                float s = 0.0f;
                for (int w = 0; w < NWAVE; ++w) s += red[w * KCLS + tid];
                bc[tid] = 1.0f / (s * (float)KCLS);
            }
            __syncthreads();
            for (int i = tid; i < NEL; i += NT) M[i] *= bc[i % KCLS];
            __syncthreads();
            for (int r = tid; r < BP; r += NT) {
                float rs = 0.0f;
                #pragma unroll
                for (int k = 0; k < KCLS; ++k) rs += M[r * KCLS + k];
                const float inv = 1.0f / (rs * (float)BP);
                #pragma unroll
                for (int k = 0; k < KCLS; ++k) M[r * KCLS + k] *= inv;
            }
            __syncthreads();
        }

        // Loss: sum over last 400 rows of (Q*800) * log_softmax(other/T).
        for (int p = tid; p < PATCH; p += NT) {
            const float* xs = Sot + p * KCLS;
            float x[KCLS];
            float xm = -3.402823466e38f;
            #pragma unroll
            for (int k = 0; k < KCLS; ++k) {
                x[k] = xs[k] * (1.0f / TEMP);
                xm = fmaxf(xm, x[k]);
            }
            float se = 0.0f;
            #pragma unroll
            for (int k = 0; k < KCLS; ++k) se += expf(x[k] - xm);
            const float lse = xm + logf(se);
            const float* q = &M[(PATCH + p) * KCLS];
            float a = 0.0f;
            #pragma unroll
            for (int k = 0; k < KCLS; ++k) a += q[k] * (x[k] - lse);
            lossAcc += a * (float)BP;
        }
        __syncthreads();   // M is rewritten by the next branch
    }

    // Deterministic block reduction -> per-sample loss.
    for (int o = 16; o > 0; o >>= 1) lossAcc += __shfl_xor(lossAcc, o, 32);
    if (lane == 0) red[wid] = lossAcc;
    __syncthreads();
    if (tid == 0) {
        float s = 0.0f;
        for (int w = 0; w < NWAVE; ++w) s += red[w];
        loss_ws[b] = -0.5f * s / (float)(PATCH * KCLS);
    }
}

__global__ void swav_finalize_kernel(const float* __restrict__ loss_ws,
                                     float* __restrict__ out)
{
    if (threadIdx.x == 0) {
        float s = 0.0f;
        for (int i = 0; i < BATCH; ++i) s += loss_ws[i];  // fixed order: deterministic
        out[0] = s / (float)BATCH;
    }
}

extern "C" void kernel_launch(void* const* d_in, const int* in_sizes, int n_in,
                              void* d_out, int out_size, void* d_ws, size_t ws_size,
                              hipStream_t stream) {
    const float* scores_t   = (const float*)d_in[0];
    const float* scores_s   = (const float*)d_in[1];
    const float* queue_base = (const float*)d_in[2];
    const float* queue_mod  = (const float*)d_in[3];
    const float* proto_w    = (const float*)d_in[4];
    float* loss_ws = (float*)d_ws;               // 256 floats of scratch

    swav_sample_kernel<<<BATCH, NT, 0, stream>>>(scores_t, scores_s,
                                                 queue_base, queue_mod,
                                                 proto_w, loss_ws);
    swav_finalize_kernel<<<1, 32, 0, stream>>>(loss_ws, (float*)d_out);
}